// DepthPredictorMultiView_73744588472386
// MI455X (gfx1250) — compile-verified
//
#include <hip/hip_runtime.h>
#include <hip/hip_bf16.h>
#include <math.h>

// Problem constants (fixed by the reference).
#define BB 8
#define VV 2
#define HH 1024
#define WW 1024
#define NN 131072
#define THMIN 0.0f
#define THMAX 500.0f
#define ALPHA_C 0.5f

typedef __attribute__((ext_vector_type(2))) float v2f;
typedef __attribute__((ext_vector_type(8))) float v8f;

struct Ws {
    float P[BB * VV * 12];     // per (b,v) 3x4 projection
    float Einv[BB * VV * 16];  // per (b,v) 4x4 inverse extrinsic
    int   nearAcc[BB * VV];    // float-bits min accumulator (+inf init)
    int   farAcc[BB * VV];     // float-bits max accumulator (-inf init)
    int   hasAcc[BB];          // any-valid flag per batch
};

// ---------------------------------------------------------------------------
// Kernel 1: per-(b,v) precompute + accumulator init (16 work items).
// ---------------------------------------------------------------------------
__global__ void dpm_setup_kernel(const float* __restrict__ intr,
                                 const float* __restrict__ extr,
                                 Ws* __restrict__ ws) {
    int t = threadIdx.x;
    if (t < BB * VV) {
        const float* E = extr + t * 16;
        float Ei[16];
        // Rigid-body inverse: [R t; 0 1]^-1 = [R^T, -R^T t; 0 1]
        #pragma unroll
        for (int i = 0; i < 3; i++)
            #pragma unroll
            for (int j = 0; j < 3; j++) Ei[i * 4 + j] = E[j * 4 + i];
        #pragma unroll
        for (int i = 0; i < 3; i++)
            Ei[i * 4 + 3] = -(E[0 * 4 + i] * E[3] + E[1 * 4 + i] * E[7] + E[2 * 4 + i] * E[11]);
        Ei[12] = 0.f; Ei[13] = 0.f; Ei[14] = 0.f; Ei[15] = 1.f;
        #pragma unroll
        for (int k = 0; k < 16; k++) ws->Einv[t * 16 + k] = Ei[k];

        const float* K = intr + t * 9;
        const float rs[3] = {(float)WW, (float)HH, 1.0f};
        #pragma unroll
        for (int i = 0; i < 3; i++)
            #pragma unroll
            for (int j = 0; j < 4; j++) {
                float s = 0.f;
                #pragma unroll
                for (int k = 0; k < 3; k++) s = fmaf(K[i * 3 + k] * rs[i], Ei[k * 4 + j], s);
                ws->P[t * 12 + i * 4 + j] = s;
            }
        ws->nearAcc[t] = 0x7f800000;  // +inf
        ws->farAcc[t]  = 0xff800000;  // -inf
    }
    if (t < BB) ws->hasAcc[t] = 0;
}

// ---------------------------------------------------------------------------
// Kernel 2: bandwidth-bound 64 MiB volume copy (B128 load/store path).
// ---------------------------------------------------------------------------
__global__ __launch_bounds__(256) void dpm_copy_kernel(const float4* __restrict__ src,
                                                       float4* __restrict__ dst, int n4) {
    int i = blockIdx.x * blockDim.x + threadIdx.x;
    int stride = gridDim.x * blockDim.x;
    for (; i < n4; i += stride) dst[i] = src[i];
}

// 3x3 minor of a 4x4 row-major matrix with row r / col c removed.
__device__ __forceinline__ float dpm_minor3(const float* g, int r, int c) {
    int rr[3], cc[3];
    int k = 0;
    #pragma unroll
    for (int i = 0; i < 4; i++) if (i != r) rr[k++] = i;
    k = 0;
    #pragma unroll
    for (int i = 0; i < 4; i++) if (i != c) cc[k++] = i;
    float a = g[rr[0] * 4 + cc[0]], b = g[rr[0] * 4 + cc[1]], c0 = g[rr[0] * 4 + cc[2]];
    float d = g[rr[1] * 4 + cc[0]], e = g[rr[1] * 4 + cc[1]], f  = g[rr[1] * 4 + cc[2]];
    float gg = g[rr[2] * 4 + cc[0]], h = g[rr[2] * 4 + cc[1]], i2 = g[rr[2] * 4 + cc[2]];
    return a * (e * i2 - f * h) - b * (d * i2 - f * gg) + c0 * (d * h - e * gg);
}

// ---------------------------------------------------------------------------
// Kernel 3: per-point triangulation. Gram matrices G = A^T A batched 4-at-a-
// time through V_WMMA_F32_16X16X4_F32 (diag 4x4 blocks of the 16x16 result),
// then adjugate-based inverse iteration for the min right-singular vector.
// ---------------------------------------------------------------------------
__global__ __launch_bounds__(256) void dpm_tri_kernel(const float* __restrict__ mk0,
                                                      const float* __restrict__ mk1,
                                                      const float* __restrict__ mconf,
                                                      const int* __restrict__ mbids,
                                                      const float* __restrict__ dispsIn,
                                                      float* __restrict__ out,
                                                      Ws* __restrict__ ws) {
    __shared__ float sA[8][32 * 16];  // per wave: 32 points x (4x4 A)
    __shared__ float sG[8][32 * 16];  // per wave: 32 points x (4x4 G)
    const int tid  = threadIdx.x;
    const int wave = tid >> 5;
    const int lane = tid & 31;
    const int gid  = blockIdx.x * 256 + tid;
    const int pid  = (gid < NN) ? gid : (NN - 1);  // clamp: keep EXEC full for WMMA

    const int   b = mbids[pid];
    const float c = mconf[pid];
    float pts[2][2];
    pts[0][0] = mk0[2 * pid]; pts[0][1] = mk0[2 * pid + 1];
    pts[1][0] = mk1[2 * pid]; pts[1][1] = mk1[2 * pid + 1];

    // Build DLT matrix A (4x4): rows [v0k0, v0k1, v1k0, v1k1], scaled by conf.
    float A[16];
    #pragma unroll
    for (int v = 0; v < 2; v++) {
        const float* P = ws->P + (b * VV + v) * 12;
        #pragma unroll
        for (int k = 0; k < 2; k++)
            #pragma unroll
            for (int j = 0; j < 4; j++)
                A[(v * 2 + k) * 4 + j] = c * (P[2 * 4 + j] * pts[v][k] - P[k * 4 + j]);
    }
    #pragma unroll
    for (int e = 0; e < 16; e++) sA[wave][lane * 16 + e] = A[e];
    __syncthreads();

    const float* wa = sA[wave];
    float* wg = sG[wave];
    // 8 groups of 4 points: D = [A0^T;..;A3^T](16x4) x [A0|..|A3](4x16).
    // Diagonal 4x4 blocks of D are the per-point Gram matrices.
    for (int g = 0; g < 8; g++) {
        const int base = g * 4;
        v2f af, bf;
        if (lane < 16) {
            // A-frag: (M=lane, K=0/1). Stacked A^T: elem(M,K) = A_p[K][i], p=M/4, i=M%4.
            int p = lane >> 2, i = lane & 3;
            af.x = wa[(base + p) * 16 + 0 * 4 + i];
            af.y = wa[(base + p) * 16 + 1 * 4 + i];
            // B-frag: (K=0/1, N=lane). Concat A: elem(K,N) = A_q[K][j], q=N/4, j=N%4.
            int q = lane >> 2, j = lane & 3;
            bf.x = wa[(base + q) * 16 + 0 * 4 + j];
            bf.y = wa[(base + q) * 16 + 1 * 4 + j];
        } else {
            int m = lane - 16;
            int p = m >> 2, i = m & 3;
            af.x = wa[(base + p) * 16 + 2 * 4 + i];  // K=2
            af.y = wa[(base + p) * 16 + 3 * 4 + i];  // K=3
            int q = m >> 2, j = m & 3;
            bf.x = wa[(base + q) * 16 + 2 * 4 + j];
            bf.y = wa[(base + q) * 16 + 3 * 4 + j];
        }
        v8f cz = {};
        v8f d = __builtin_amdgcn_wmma_f32_16x16x4_f32(false, af, false, bf,
                                                      (short)0, cz, false, false);
        // Extract diag blocks. Lane L<16 holds col N=L rows 0..7 in d[0..7];
        // lane L>=16 holds col N=L-16 rows 8..15 in d[0..7].
        if (lane < 8) {
            int p = lane >> 2, j = lane & 3;  // G_0 (lanes 0-3), G_1 (lanes 4-7)
            float e0, e1, e2, e3;
            if (p == 0) { e0 = d[0]; e1 = d[1]; e2 = d[2]; e3 = d[3]; }
            else        { e0 = d[4]; e1 = d[5]; e2 = d[6]; e3 = d[7]; }
            wg[(base + p) * 16 + 0 * 4 + j] = e0;
            wg[(base + p) * 16 + 1 * 4 + j] = e1;
            wg[(base + p) * 16 + 2 * 4 + j] = e2;
            wg[(base + p) * 16 + 3 * 4 + j] = e3;
        } else if (lane >= 24) {
            int n = lane - 16;                // 8..15 -> G_2 / G_3
            int p = n >> 2, j = n & 3;        // p = 2 or 3
            float e0, e1, e2, e3;
            if (p == 2) { e0 = d[0]; e1 = d[1]; e2 = d[2]; e3 = d[3]; }
            else        { e0 = d[4]; e1 = d[5]; e2 = d[6]; e3 = d[7]; }
            wg[(base + p) * 16 + 0 * 4 + j] = e0;
            wg[(base + p) * 16 + 1 * 4 + j] = e1;
            wg[(base + p) * 16 + 2 * 4 + j] = e2;
            wg[(base + p) * 16 + 3 * 4 + j] = e3;
        }
    }
    __syncthreads();

    // Each lane solves its own point: min-eigenvector of its 4x4 Gram via
    // adjugate-based inverse iteration (adj(G) ~ det(G) G^-1).
    float Gm[16];
    #pragma unroll
    for (int e = 0; e < 16; e++) Gm[e] = wg[lane * 16 + e];

    float adj[16];
    #pragma unroll
    for (int i = 0; i < 4; i++)
        #pragma unroll
        for (int j = 0; j < 4; j++) {
            float m = dpm_minor3(Gm, j, i);
            adj[i * 4 + j] = (((i + j) & 1) ? -m : m);
        }

    float v4[4] = {1.f, 1.f, 1.f, 1.f};
    #pragma unroll
    for (int it = 0; it < 3; it++) {
        float r[4];
        #pragma unroll
        for (int i = 0; i < 4; i++)
            r[i] = fmaf(adj[i * 4 + 0], v4[0],
                   fmaf(adj[i * 4 + 1], v4[1],
                   fmaf(adj[i * 4 + 2], v4[2], adj[i * 4 + 3] * v4[3])));
        float n2 = r[0] * r[0] + r[1] * r[1] + r[2] * r[2] + r[3] * r[3];
        if (n2 > 1e-30f) {
            float inv = rsqrtf(n2);
            #pragma unroll
            for (int i = 0; i < 4; i++) v4[i] = r[i] * inv;
        }
    }

    const float wv = v4[3];
    const float px = v4[0] / wv, py = v4[1] / wv, pz = v4[2] / wv;

    float z[2];
    bool valid = true;
    #pragma unroll
    for (int v = 0; v < 2; v++) {
        const float* Ei = ws->Einv + (b * VV + v) * 16;
        z[v] = fmaf(Ei[8], px, fmaf(Ei[9], py, fmaf(Ei[10], pz, Ei[11])));
        valid = valid && (z[v] > THMIN) && (z[v] < THMAX);
    }

    const bool active = (gid < NN);
    if (active && valid) {
        #pragma unroll
        for (int v = 0; v < 2; v++) {
            // z > 0: positive-float bits order == int order.
            atomicMin(&ws->nearAcc[b * VV + v], __float_as_int(z[v]));
            atomicMax(&ws->farAcc[b * VV + v], __float_as_int(z[v]));
        }
        atomicOr(&ws->hasAcc[b], 1);
    }

    if (active) {
        #pragma unroll
        for (int v = 0; v < 2; v++) {
            int x = (int)pts[v][0]; x = min(max(x, 0), WW - 1);
            int y = (int)pts[v][1]; y = min(max(y, 0), HH - 1);
            size_t idx = ((size_t)(v * BB + b) * HH + (size_t)y) * WW + (size_t)x;
            float cost = dispsIn[idx];
            float val = valid ? (ALPHA_C * (1.0f / z[v]) + (1.0f - ALPHA_C) * cost) : cost;
            out[idx] = val;
        }
    }
}

// ---------------------------------------------------------------------------
// Kernel 4: finalize near/far/flag (B x V each) into the output tail.
// ---------------------------------------------------------------------------
__global__ void dpm_fin_kernel(const Ws* __restrict__ ws, float* __restrict__ out) {
    int t = threadIdx.x;
    if (t < BB * VV) {
        int b = t / VV;
        int has = ws->hasAcc[b];
        float nearv = has ? __int_as_float(ws->nearAcc[t]) : 0.0f;
        float farv  = has ? __int_as_float(ws->farAcc[t])  : 500.0f;
        float flagv = has ? 1.0f : 0.0f;
        size_t baseo = (size_t)VV * BB * HH * WW;
        out[baseo + t]                 = nearv;
        out[baseo + BB * VV + t]       = farv;
        out[baseo + 2 * BB * VV + t]   = flagv;
    }
}

extern "C" void kernel_launch(void* const* d_in, const int* in_sizes, int n_in,
                              void* d_out, int out_size, void* d_ws, size_t ws_size,
                              hipStream_t stream) {
    const float* mk0   = (const float*)d_in[0];
    const float* mk1   = (const float*)d_in[1];
    const float* mconf = (const float*)d_in[2];
    const int*   mbids = (const int*)d_in[3];
    const float* intr  = (const float*)d_in[4];
    const float* extr  = (const float*)d_in[5];
    const float* disps = (const float*)d_in[6];
    float* out = (float*)d_out;
    Ws* ws = (Ws*)d_ws;

    dpm_setup_kernel<<<1, 32, 0, stream>>>(intr, extr, ws);

    const int n4 = (VV * BB * HH * WW) / 4;  // 4M float4 = 64 MiB
    dpm_copy_kernel<<<2048, 256, 0, stream>>>((const float4*)disps, (float4*)out, n4);

    dpm_tri_kernel<<<NN / 256, 256, 0, stream>>>(mk0, mk1, mconf, mbids, disps, out, ws);

    dpm_fin_kernel<<<1, 32, 0, stream>>>(ws, out);
}